// TensorRelaxationPooling_34213709480472
// MI455X (gfx1250) — compile-verified
//
#include <hip/hip_runtime.h>

typedef __attribute__((ext_vector_type(2))) float v2f;
typedef __attribute__((ext_vector_type(4))) float v4f;
typedef __attribute__((ext_vector_type(8))) float v8f;

// D = A(16x4) * B(4x16) + C, f32 WMMA (CDNA5). 8-arg form:
// (neg_a, A, neg_b, B, c_mod, C, reuse_a, reuse_b)
#define WMMA_F32(a, b, c)                                                     \
  __builtin_amdgcn_wmma_f32_16x16x4_f32(false, (a), false, (b), (short)0,     \
                                        (c), false, false)

// B=8, T=4096, D_IN=128, NB=32, STRIDE=2.
// One wave (32 threads) per block. Each block handles 16 raw frames
// (= 8 pooled output frames). Grid = B*T/16 = 2048 blocks.
__global__ __launch_bounds__(32) void trp_wmma_kernel(
    const float* __restrict__ X,   // [B*T, 128]
    const float* __restrict__ E,   // [128, 32]
    const float* __restrict__ F,   // [128, 32]
    const float* __restrict__ G,   // [1]
    const float* __restrict__ bx,  // [32]
    const float* __restrict__ by,  // [32]
    float* __restrict__ out)       // [B*T/2, 1024]
{
  __shared__ __align__(16) float lz1[16 * 32];   // z1 tile, row-major [frame][n]
  __shared__ __align__(16) float lz2[16 * 32];   // z2 tile
  __shared__ __align__(16) float lout[32 * 32];  // one pooled 32x32 out tile

  const int l  = threadIdx.x;  // 0..31
  const int lh = l & 15;       // lane%16
  const int hi = l >> 4;       // 0: lanes 0-15, 1: lanes 16-31

  const int frame0 = blockIdx.x * 16;  // flat raw-frame base over B*T

  // ---- Phase 1: Z1 = X*E, Z2 = X*F via f32 WMMA, K=128 in 32 steps ----
  // A-frag (16x4 f32): lane lh holds row M=lh; VGPR0/1 = K = 2*hi, 2*hi+1.
  const float* xbase = X + (size_t)(frame0 + lh) * 128 + 2 * hi;
  // B-frag (4x16 f32): VGPR0 = rows K=0 (lanes 0-15) / K=2 (lanes 16-31),
  //                    VGPR1 = rows K=1 / K=3.  E row stride = 32.
  const float* ebase = E + hi * 64 + lh;
  const float* fbase = F + hi * 64 + lh;

  v8f z1a = {}, z1b = {}, z2a = {}, z2b = {};  // n-tiles 0 and 1 for E and F
  #pragma unroll
  for (int s = 0; s < 32; ++s) {
    v2f a = *(const v2f*)(xbase + 4 * s);
    v2f be0, be1, bf0, bf1;
    be0.x = ebase[s * 128];      be0.y = ebase[s * 128 + 32];
    be1.x = ebase[s * 128 + 16]; be1.y = ebase[s * 128 + 48];
    bf0.x = fbase[s * 128];      bf0.y = fbase[s * 128 + 32];
    bf1.x = fbase[s * 128 + 16]; bf1.y = fbase[s * 128 + 48];
    z1a = WMMA_F32(a, be0, z1a);
    z1b = WMMA_F32(a, be1, z1b);
    z2a = WMMA_F32(a, bf0, z2a);
    z2b = WMMA_F32(a, bf1, z2b);
  }

  // ---- Stage z to LDS, folding bias and the G/2 pool scale into z1 ----
  const float g   = 0.5f * G[0];
  const float bx0 = bx[lh], bx1 = bx[lh + 16];
  const float by0 = by[lh], by1 = by[lh + 16];
  // C/D layout: VGPR v, lane -> (M = v + 8*hi, N = lh) within a 16x16 tile.
  #pragma unroll
  for (int v = 0; v < 8; ++v) {
    const int row = (v + 8 * hi) * 32 + lh;   // [frame][n]
    lz1[row]      = (z1a[v] + bx0) * g;
    lz1[row + 16] = (z1b[v] + bx1) * g;
    lz2[row]      =  z2a[v] + by0;
    lz2[row + 16] =  z2b[v] + by1;
  }
  __syncthreads();

  float* outb = out + (size_t)blockIdx.x * 8 * 1024;
  const float msk = (hi == 0) ? 1.0f : 0.0f;  // zero K=2,3 rows (branch-free)

  // ---- Phase 2: per pooled frame p, out = Z1^T(32x2) * Z2(2x32) ----
  #pragma unroll
  for (int p = 0; p < 8; ++p) {
    const int r0 = (2 * p) * 32 + lh;
    const int r1 = (2 * p + 1) * 32 + lh;
    v2f a0, a1, b0, b1;
    a0.x = lz1[r0] * msk;      a0.y = lz1[r1] * msk;       // rows n1 = 0..15
    a1.x = lz1[r0 + 16] * msk; a1.y = lz1[r1 + 16] * msk;  // rows n1 = 16..31
    b0.x = lz2[r0] * msk;      b0.y = lz2[r1] * msk;       // cols n2 = 0..15
    b1.x = lz2[r0 + 16] * msk; b1.y = lz2[r1 + 16] * msk;  // cols n2 = 16..31

    v8f c = {};
    v8f d00 = WMMA_F32(a0, b0, c);
    v8f d01 = WMMA_F32(a0, b1, c);
    v8f d10 = WMMA_F32(a1, b0, c);
    v8f d11 = WMMA_F32(a1, b1, c);

    // Stage 32x32 tile to LDS row-major (undo accumulator striping).
    #pragma unroll
    for (int v = 0; v < 8; ++v) {
      const int rr = (v + 8 * hi) * 32 + lh;   // (n1)*32 + n2
      lout[rr]            = d00[v];
      lout[rr + 16]       = d01[v];
      lout[rr + 512]      = d10[v];            // n1 + 16 -> +16*32
      lout[rr + 512 + 16] = d11[v];
    }
    __syncthreads();

    // Coalesced streaming store: 8 x (32 lanes x 16B) = 4KB contiguous.
    float* op = outb + p * 1024;
    #pragma unroll
    for (int s2 = 0; s2 < 8; ++s2) {
      const int ofs = (s2 * 32 + l) * 4;
      v4f vv = *(const v4f*)&lout[ofs];
      __builtin_nontemporal_store(vv, (v4f*)(op + ofs));
    }
    __syncthreads();  // lout reused next iteration
  }
}

extern "C" void kernel_launch(void* const* d_in, const int* in_sizes, int n_in,
                              void* d_out, int out_size, void* d_ws, size_t ws_size,
                              hipStream_t stream) {
  const float* X  = (const float*)d_in[0];
  const float* E  = (const float*)d_in[1];
  const float* F  = (const float*)d_in[2];
  const float* G  = (const float*)d_in[3];
  const float* bx = (const float*)d_in[4];
  const float* by = (const float*)d_in[5];
  float* out = (float*)d_out;

  const int nframes = in_sizes[0] / 128;  // B*T = 32768
  const int blocks  = nframes / 16;       // 2048
  trp_wmma_kernel<<<blocks, 32, 0, stream>>>(X, E, F, G, bx, by, out);
}